// MaxPool3dMPS_43258910605942
// MI455X (gfx1250) — compile-verified
//
#include <hip/hip_runtime.h>

// 3D max pool, kernel=2, stride=2, pad=0.
// x: [2, 32, 128, 128, 128] f32  ->  y: [2, 32, 64, 64, 64] f32
//
// Bandwidth-bound (576 MB traffic, ~24.7 us floor at 23.3 TB/s).
// One wave32 per output row: lane loads float4 from each of the 4 input rows
// of its 2x2x2 windows (all loads 512B-contiguous per wave -> global_load_b128),
// reduces with v_max_num_f32, stores float2 (global_store_b64).
// Non-temporal hints: input (512MB) streams once and exceeds the 192MB L2.

typedef float v4f __attribute__((ext_vector_type(4)));
typedef float v2f __attribute__((ext_vector_type(2)));

#define IN_W   128
#define IN_HW  (128 * 128)          // 16384 floats per (d) plane
#define IN_DHW (128 * 128 * 128)    // floats per (b,c) volume
#define OUT_W   64
#define OUT_HW  (64 * 64)
#define OUT_DHW (64 * 64 * 64)

__global__ __launch_bounds__(256) void maxpool3d_k2s2_kernel(
    const float* __restrict__ x, float* __restrict__ y)
{
    unsigned tid = blockIdx.x * blockDim.x + threadIdx.x;

    // 32 lane-slots per output row (64 outputs, 2 per lane).
    unsigned slot = tid & 31u;        // which float4 chunk of the input row
    unsigned row  = tid >> 5;         // output row id in [0, 2*32*64*64)
    unsigned oh   = row & 63u;
    unsigned t    = row >> 6;
    unsigned od   = t & 63u;
    unsigned bc   = t >> 6;           // fused (b,c) in [0, 64)

    // Input base: (b,c) volume + d=2*od plane + h=2*oh row + lane chunk.
    size_t in_base = (size_t)bc * IN_DHW
                   + (size_t)(2u * od) * IN_HW
                   + (size_t)(2u * oh) * IN_W
                   + (size_t)slot * 4u;

    const v4f* p00 = (const v4f*)(x + in_base);                  // d0,h0
    const v4f* p01 = (const v4f*)(x + in_base + IN_W);           // d0,h1
    const v4f* p10 = (const v4f*)(x + in_base + IN_HW);          // d1,h0
    const v4f* p11 = (const v4f*)(x + in_base + IN_HW + IN_W);   // d1,h1

    v4f a = __builtin_nontemporal_load(p00);
    v4f b = __builtin_nontemporal_load(p01);
    v4f c = __builtin_nontemporal_load(p10);
    v4f d = __builtin_nontemporal_load(p11);

    // Vertical reduction across the 4 input rows (component-wise).
    v4f m;
    m.x = fmaxf(fmaxf(a.x, b.x), fmaxf(c.x, d.x));
    m.y = fmaxf(fmaxf(a.y, b.y), fmaxf(c.y, d.y));
    m.z = fmaxf(fmaxf(a.z, b.z), fmaxf(c.z, d.z));
    m.w = fmaxf(fmaxf(a.w, b.w), fmaxf(c.w, d.w));

    // Horizontal reduction along W: pairs (x,y) and (z,w) -> 2 outputs.
    v2f o;
    o.x = fmaxf(m.x, m.y);
    o.y = fmaxf(m.z, m.w);

    size_t out_base = (size_t)bc * OUT_DHW
                    + (size_t)od * OUT_HW
                    + (size_t)oh * OUT_W
                    + (size_t)slot * 2u;

    __builtin_nontemporal_store(o, (v2f*)(y + out_base));
}

extern "C" void kernel_launch(void* const* d_in, const int* in_sizes, int n_in,
                              void* d_out, int out_size, void* d_ws, size_t ws_size,
                              hipStream_t stream)
{
    const float* x = (const float*)d_in[0];
    float*       y = (float*)d_out;

    // Total threads: 2*32*64*64 rows * 32 lanes = 8,388,608
    const unsigned total_threads = 2u * 32u * 64u * 64u * 32u;
    const unsigned block = 256u;                 // 8 wave32s per block
    const unsigned grid  = total_threads / block; // 32768, exact

    maxpool3d_k2s2_kernel<<<grid, block, 0, stream>>>(x, y);
}